// MultiHeadLatentAttention_22351009809309
// MI455X (gfx1250) — compile-verified
//
#include <hip/hip_runtime.h>
#include <math.h>

#define NUM_HEADS 16
#define HIDDEN    2048
#define HEAD_DIM  128
#define LATENT    64
#define BLK       64
#define BATCH     2
#define SEQ       2048

typedef float v2f __attribute__((ext_vector_type(2)));
typedef float v8f __attribute__((ext_vector_type(8)));

__device__ __forceinline__ v8f wmma_f32(v2f a, v2f b, v8f c) {
    // D = A(16x4,f32) * B(4x16,f32) + C(16x16,f32)
    return __builtin_amdgcn_wmma_f32_16x16x4_f32(
        false, a, false, b, (short)0, c, false, false);
}

// ---------------------------------------------------------------------------
// Kernel 1: per-head Q/K/V projections.
// One wave computes a 16(seq) x (NT*16) strip for one tensor of one head:
// the A (activation) float2 load is reused across all NT column tiles, giving
// NT independent WMMA chains per K-step.
// which: 0 -> q (NT=4), 1 -> k (NT=4), 2 -> v (NT=8).
// ---------------------------------------------------------------------------
template<int NT, int LDW>
__device__ __forceinline__ void qkv_strip(
    const float* __restrict__ arow, const float* __restrict__ W,
    const float* __restrict__ bias, float* __restrict__ out,
    int b, int s0, int h, int g, int m)
{
    v8f acc[NT];
#pragma unroll
    for (int nt = 0; nt < NT; ++nt) acc[nt] = (v8f){0.f,0.f,0.f,0.f,0.f,0.f,0.f,0.f};

    for (int k0 = 0; k0 < HEAD_DIM; k0 += 4) {
        v2f a;
        a.x = arow[k0 + 2*g + 0];
        a.y = arow[k0 + 2*g + 1];
        const float* wrow0 = W + (size_t)(k0 + 2*g + 0) * LDW + m;
        const float* wrow1 = W + (size_t)(k0 + 2*g + 1) * LDW + m;
#pragma unroll
        for (int nt = 0; nt < NT; ++nt) {
            v2f bm;
            bm.x = wrow0[nt * 16];
            bm.y = wrow1[nt * 16];
            acc[nt] = wmma_f32(a, bm, acc[nt]);
        }
    }
#pragma unroll
    for (int nt = 0; nt < NT; ++nt) {
        float bb = bias[nt * 16 + m];
#pragma unroll
        for (int v = 0; v < 8; ++v) {
            int s = s0 + v + 8 * g;
            out[(((size_t)(b * SEQ + s)) * NUM_HEADS + h) * (size_t)LDW + nt * 16 + m] =
                acc[nt][v] + bb;
        }
    }
}

__global__ __launch_bounds__(32) void qkv_kernel(
    const float* __restrict__ hs,
    const float* __restrict__ Wq, const float* __restrict__ bq,
    const float* __restrict__ Wk, const float* __restrict__ bk,
    const float* __restrict__ Wv, const float* __restrict__ bv,
    float* __restrict__ qbuf, float* __restrict__ kbuf, float* __restrict__ vbuf)
{
    int wid   = blockIdx.x;
    int which = wid % 3;
    int rest  = wid / 3;
    int qt = rest % (SEQ / 16);
    int bh = rest / (SEQ / 16);
    int h  = bh % NUM_HEADS;
    int b  = bh / NUM_HEADS;

    int lane = threadIdx.x;
    int g = lane >> 4, m = lane & 15;
    int s0 = qt * 16;

    // A row m = sequence position s0+m, K dim = head_dim (128)
    const float* arow = hs + ((size_t)(b * SEQ + s0 + m)) * HIDDEN + h * HEAD_DIM;

    if (which == 0)
        qkv_strip<4, LATENT>(arow, Wq, bq, qbuf, b, s0, h, g, m);
    else if (which == 1)
        qkv_strip<4, LATENT>(arow, Wk, bk, kbuf, b, s0, h, g, m);
    else
        qkv_strip<8, HEAD_DIM>(arow, Wv, bv, vbuf, b, s0, h, g, m);
}

// ---------------------------------------------------------------------------
// Kernel 2: RoPE in place on q and k.  Pair i in [0,32): angle = s * 10000^(-((4i)%64)/64)
// ---------------------------------------------------------------------------
__global__ void rope_kernel(float* __restrict__ qbuf, float* __restrict__ kbuf, int n)
{
    int idx = blockIdx.x * blockDim.x + threadIdx.x;
    if (idx >= n) return;
    int i = idx & 31;
    int t = idx >> 5;
    int h = t % NUM_HEADS; t /= NUM_HEADS;
    int s = t % SEQ;
    int b = t / SEQ;

    float expo = (float)((4 * i) & 63) / 64.0f;
    float invf = powf(10000.0f, -expo);
    float ang  = (float)s * invf;
    float c  = cosf(ang);
    float sn = sinf(ang);

    size_t base = ((((size_t)(b * SEQ + s)) * NUM_HEADS + h) * LATENT) + 2 * i;
    float xe = qbuf[base], xo = qbuf[base + 1];
    qbuf[base]     = xe * c  - xo * sn;
    qbuf[base + 1] = xe * sn + xo * c;
    xe = kbuf[base]; xo = kbuf[base + 1];
    kbuf[base]     = xe * c  - xo * sn;
    kbuf[base + 1] = xe * sn + xo * c;
}

// ---------------------------------------------------------------------------
// Kernel 3: attention. One wave handles 16 queries of one (b,h). Block-local
// softmax means each 64-key block is independent: compute 16x64 scores (WMMA),
// softmax rows over 64 cols (shfl within half-wave), round-trip P through LDS
// into A-layout, then ctx += P * V (WMMA).
// ---------------------------------------------------------------------------
__global__ __launch_bounds__(32) void attn_kernel(
    const float* __restrict__ qbuf, const float* __restrict__ kbuf,
    const float* __restrict__ vbuf, const float* __restrict__ amask,
    float* __restrict__ ctx)
{
    __shared__ float q_lds[16 * 64];
    __shared__ float p_lds[16 * 68];   // stride 68 to dodge bank conflicts

    int wid = blockIdx.x;
    int qt = wid % (SEQ / 16);
    int bh = wid / (SEQ / 16);
    int h  = bh % NUM_HEADS;
    int b  = bh / NUM_HEADS;

    int lane = threadIdx.x;
    int g = lane >> 4, m = lane & 15;
    int q0 = qt * 16;
    int qB = q0 / BLK;                  // all 16 queries share one key-block index

    // stage q tile (16x64) in LDS, then pre-pack A-layout registers
    for (int t = lane; t < 16 * 64; t += 32) {
        int r = t >> 6, cc = t & 63;
        q_lds[t] = qbuf[(((size_t)(b * SEQ + q0 + r)) * NUM_HEADS + h) * LATENT + cc];
    }
    __syncthreads();
    v2f areg[16];
#pragma unroll
    for (int k0 = 0; k0 < 16; ++k0) {
        areg[k0].x = q_lds[m * 64 + k0 * 4 + 2 * g + 0];
        areg[k0].y = q_lds[m * 64 + k0 * 4 + 2 * g + 1];
    }

    v8f acc[8];
#pragma unroll
    for (int i = 0; i < 8; ++i) acc[i] = (v8f){0.f,0.f,0.f,0.f,0.f,0.f,0.f,0.f};

    for (int kb = 0; kb <= qB; ++kb) {
        // ---- scores: 4 subtiles of 16 keys, K = latent 64 ----
        v8f sc[4];
#pragma unroll
        for (int st = 0; st < 4; ++st) {
            v8f c = {0.f,0.f,0.f,0.f,0.f,0.f,0.f,0.f};
            int key = kb * BLK + st * 16 + m;               // B col n = m
            const float* krow = kbuf + (((size_t)(b * SEQ + key)) * NUM_HEADS + h) * LATENT;
#pragma unroll
            for (int k0 = 0; k0 < 16; ++k0) {
                v2f bv2;
                bv2.x = krow[k0 * 4 + 2 * g + 0];
                bv2.y = krow[k0 * 4 + 2 * g + 1];
                c = wmma_f32(areg[k0], bv2, c);
            }
            float am = amask[(size_t)b * SEQ + key];
#pragma unroll
            for (int v = 0; v < 8; ++v) {
                int qrow = q0 + v + 8 * g;
                float val = c[v] * 0.125f;                  // 1/sqrt(64)
                bool ok = (key <= qrow) && (am != 0.0f);
                c[v] = ok ? val : -1.0e9f;
            }
            sc[st] = c;
        }

        // ---- row softmax over the 64 keys of this block ----
#pragma unroll
        for (int v = 0; v < 8; ++v) {
            float mx = fmaxf(fmaxf(sc[0][v], sc[1][v]), fmaxf(sc[2][v], sc[3][v]));
            for (int off = 1; off < 16; off <<= 1)
                mx = fmaxf(mx, __shfl_xor(mx, off, 16));
            float e0 = __expf(sc[0][v] - mx);
            float e1 = __expf(sc[1][v] - mx);
            float e2 = __expf(sc[2][v] - mx);
            float e3 = __expf(sc[3][v] - mx);
            float sm = e0 + e1 + e2 + e3;
            for (int off = 1; off < 16; off <<= 1)
                sm += __shfl_xor(sm, off, 16);
            float inv = 1.0f / sm;
            sc[0][v] = e0 * inv; sc[1][v] = e1 * inv;
            sc[2][v] = e2 * inv; sc[3][v] = e3 * inv;
        }

        // ---- P (C-layout) -> LDS -> A-layout ----
#pragma unroll
        for (int st = 0; st < 4; ++st)
#pragma unroll
            for (int v = 0; v < 8; ++v)
                p_lds[(v + 8 * g) * 68 + st * 16 + m] = sc[st][v];
        __syncthreads();

        v2f preg[16];
#pragma unroll
        for (int k0 = 0; k0 < 16; ++k0) {
            preg[k0].x = p_lds[m * 68 + k0 * 4 + 2 * g + 0];
            preg[k0].y = p_lds[m * 68 + k0 * 4 + 2 * g + 1];
        }

        // ---- ctx(16x128) += P(16x64) * V(64x128) ----
#pragma unroll
        for (int nt = 0; nt < 8; ++nt) {
            v8f a2 = acc[nt];
#pragma unroll
            for (int k0 = 0; k0 < 16; ++k0) {
                int key0 = kb * BLK + k0 * 4 + 2 * g;
                v2f bv2;
                bv2.x = vbuf[(((size_t)(b * SEQ + key0 + 0)) * NUM_HEADS + h) * HEAD_DIM + nt * 16 + m];
                bv2.y = vbuf[(((size_t)(b * SEQ + key0 + 1)) * NUM_HEADS + h) * HEAD_DIM + nt * 16 + m];
                a2 = wmma_f32(preg[k0], bv2, a2);
            }
            acc[nt] = a2;
        }
        __syncthreads();   // p_lds reused next kb
    }

#pragma unroll
    for (int nt = 0; nt < 8; ++nt)
#pragma unroll
        for (int v = 0; v < 8; ++v) {
            int srow = q0 + v + 8 * g;
            ctx[((size_t)(b * SEQ + srow)) * HIDDEN + h * HEAD_DIM + nt * 16 + m] = acc[nt][v];
        }
}

// ---------------------------------------------------------------------------
// Kernel 4: output projection  out = ctx(4096x2048) @ Wo(2048x2048) + bo
// One wave per 16x64 output strip: A float2 reused across 4 column tiles,
// 4 independent WMMA chains per K-step.
// ---------------------------------------------------------------------------
__global__ __launch_bounds__(32) void proj_kernel(
    const float* __restrict__ ctx, const float* __restrict__ Wo,
    const float* __restrict__ bo, float* __restrict__ out)
{
    int wid = blockIdx.x;
    int cs = wid % (HIDDEN / 64);       // 64-wide column strip
    int rt = wid / (HIDDEN / 64);
    int lane = threadIdx.x;
    int g = lane >> 4, m = lane & 15;
    int r0 = rt * 16, n0 = cs * 64;

    const float* arow = ctx + (size_t)(r0 + m) * HIDDEN;
    v8f acc[4];
#pragma unroll
    for (int nt = 0; nt < 4; ++nt) acc[nt] = (v8f){0.f,0.f,0.f,0.f,0.f,0.f,0.f,0.f};

    for (int k0 = 0; k0 < HIDDEN; k0 += 4) {
        v2f a;
        a.x = arow[k0 + 2 * g + 0];
        a.y = arow[k0 + 2 * g + 1];
        const float* wrow0 = Wo + (size_t)(k0 + 2 * g + 0) * HIDDEN + n0 + m;
        const float* wrow1 = Wo + (size_t)(k0 + 2 * g + 1) * HIDDEN + n0 + m;
#pragma unroll
        for (int nt = 0; nt < 4; ++nt) {
            v2f bm;
            bm.x = wrow0[nt * 16];
            bm.y = wrow1[nt * 16];
            acc[nt] = wmma_f32(a, bm, acc[nt]);
        }
    }
#pragma unroll
    for (int nt = 0; nt < 4; ++nt) {
        float bb = bo[n0 + nt * 16 + m];
#pragma unroll
        for (int v = 0; v < 8; ++v)
            out[(size_t)(r0 + v + 8 * g) * HIDDEN + n0 + nt * 16 + m] = acc[nt][v] + bb;
    }
}

// ---------------------------------------------------------------------------
extern "C" void kernel_launch(void* const* d_in, const int* in_sizes, int n_in,
                              void* d_out, int out_size, void* d_ws, size_t ws_size,
                              hipStream_t stream)
{
    const float* hs    = (const float*)d_in[0];
    const float* amask = (const float*)d_in[1];
    const float* Wq    = (const float*)d_in[2];
    const float* bq    = (const float*)d_in[3];
    const float* Wk    = (const float*)d_in[4];
    const float* bk    = (const float*)d_in[5];
    const float* Wv    = (const float*)d_in[6];
    const float* bv    = (const float*)d_in[7];
    const float* Wo    = (const float*)d_in[8];
    const float* bo    = (const float*)d_in[9];
    float* out = (float*)d_out;

    // workspace: q(16MB) k(16MB) v(32MB) ctx(32MB) = 96MB
    float* qbuf = (float*)d_ws;
    float* kbuf = qbuf + (size_t)BATCH * SEQ * NUM_HEADS * LATENT;
    float* vbuf = kbuf + (size_t)BATCH * SEQ * NUM_HEADS * LATENT;
    float* ctxb = vbuf + (size_t)BATCH * SEQ * NUM_HEADS * HEAD_DIM;

    // 1) QKV projections: B*H*(S/16) seq-tiles * 3 tensors
    int qkv_blocks = BATCH * NUM_HEADS * (SEQ / 16) * 3;
    qkv_kernel<<<qkv_blocks, 32, 0, stream>>>(hs, Wq, bq, Wk, bk, Wv, bv, qbuf, kbuf, vbuf);

    // 2) RoPE
    int nrope = BATCH * SEQ * NUM_HEADS * 32;
    rope_kernel<<<(nrope + 255) / 256, 256, 0, stream>>>(qbuf, kbuf, nrope);

    // 3) attention
    int attn_blocks = BATCH * NUM_HEADS * (SEQ / 16);
    attn_kernel<<<attn_blocks, 32, 0, stream>>>(qbuf, kbuf, vbuf, amask, ctxb);

    // 4) output projection: 16x64 strips
    int proj_blocks = (BATCH * SEQ / 16) * (HIDDEN / 64);
    proj_kernel<<<proj_blocks, 32, 0, stream>>>(ctxb, Wo, bo, out);
}